// Piecewise_22780506538397
// MI455X (gfx1250) — compile-verified
//
#include <hip/hip_runtime.h>
#include <stdint.h>

// Piecewise-Lagrange (KAN) layer as a dense fp32 WMMA GEMM.
//   out[b,o] = sum_{kg} A[b,kg] * Wflat[o,kg],  kg = i*97 + kk,  Ktot = 99328
// A has exactly 4 nonzeros per (b,i) (the Lagrange basis window); densifying
// with 0.0f is exact in fp32. W streams once from HBM (406 MB -> ~17.5us floor
// at 23.3 TB/s); A (12.7 MB) stays L2-resident.

#define IN_F    1024
#define OUT_F   1024
#define BATCH   32
#define KC      97
#define KTOT    (IN_F * KC)          // 99328
#define NCHUNK  32                   // k split across waves for occupancy
#define KCHUNK  (KTOT / NCHUNK)      // 3104 (= 32 input features)
#define KSTEP   32                   // k per LDS stage
#define NSTAGE  (KCHUNK / KSTEP)     // 97
#define LDS_ROW 36                   // padded row stride (floats): 16B-aligned, bank-conflict-free
#define TILE_FLOATS (16 * LDS_ROW)   // one W tile: 16 o-rows x 32 k (+pad)
#define TILE_BYTES  (TILE_FLOATS * 4)

typedef float v2f __attribute__((ext_vector_type(2)));
typedef float v8f __attribute__((ext_vector_type(8)));

// ---------------------------------------------------------------------------
// Kernel 1: build transposed, pair-interleaved dense basis matrix
//   ATp[f], f = kp*64 + row*2 + c  holds A[k = 2*kp + c, row=b]
// so a WMMA A-fragment (16x4 f32) is one coalesced b64 load per lane.
// ---------------------------------------------------------------------------
__global__ void build_A(const float* __restrict__ x, float* __restrict__ atp) {
    int f  = blockIdx.x * blockDim.x + threadIdx.x;   // < KTOT*BATCH
    int kp = f >> 6;
    int r  = f & 63;
    int b  = r >> 1;
    int c  = r & 1;
    int k  = kp * 2 + c;
    int i  = k / KC;
    int kk = k - i * KC;

    float xv = x[b * IN_F + i];
    float t  = (xv + 1.0f) * 16.0f;       // (x+half)/length*segments
    int id   = (int)t;                    // trunc == floor for t>=0
    id = id < 0 ? 0 : (id > 31 ? 31 : id);
    int j = kk - id * 3;                  // node index within window, if any
    float val = 0.0f;
    if (j >= 0 && j < 4) {
        // local coord on [-1,1]; nodes X = {-1,-1/2,1/2,1} (Chebyshev-Lobatto, N=4)
        float xin = 32.0f * (xv - ((float)id * 0.0625f - 1.0f)) - 1.0f;
        float d0 = xin + 1.0f;
        float d1 = xin + 0.5f;
        float d2 = xin - 0.5f;
        float d3 = xin - 1.0f;
        switch (j) {                       // basis_j = prod_{m!=j}(x-X_m)/denom_j
            case 0:  val = d1 * d2 * d3 * (-0.66666667f); break;  // denom -1.5
            case 1:  val = d0 * d2 * d3 * ( 1.33333333f); break;  // denom  0.75
            case 2:  val = d0 * d1 * d3 * (-1.33333333f); break;  // denom -0.75
            default: val = d0 * d1 * d2 * ( 0.66666667f); break;  // denom  1.5
        }
    }
    atp[f] = val;
}

// ---------------------------------------------------------------------------
// Kernel 2: per-(o-tile, k-chunk) partial GEMM with WMMA f32 16x16x4.
// One wave per workgroup: it owns its LDS double-buffer, so only wave-local
// counter waits (s_wait_asynccnt) are needed -- no barriers.
// W tile (16 o-rows x 32 k) is staged HBM->LDS with async-to-LDS b128.
// ---------------------------------------------------------------------------
__global__ void __launch_bounds__(32) gemm_part(const float* __restrict__ wmat,
                                                const float* __restrict__ atp,
                                                float* __restrict__ pw) {
    __shared__ float lds[2 * TILE_FLOATS];

    const int L      = threadIdx.x;
    const int otile  = blockIdx.x & 63;          // 64 o-tiles of 16
    const int chunk  = blockIdx.x >> 6;          // 32 k-chunks
    const int kbase  = chunk * KCHUNK;
    const int lane15 = L & 15;
    const int half   = L >> 4;

    // W row o is a contiguous 99328-float run: Wflat[o*KTOT + kg]
    const float* wbase = wmat + (size_t)(otile * 16) * KTOT + kbase;
    const unsigned lds_base = (unsigned)(uintptr_t)(&lds[0]);

    // async stage loader: 128 b128 chunks (16 rows x 128B), 4 insts/lane
    auto stage_load = [&](int s, int p) {
        const float* g0 = wbase + s * KSTEP + (size_t)(L >> 3) * KTOT + (L & 7) * 4;
        unsigned lo = lds_base + (unsigned)p * TILE_BYTES
                    + (unsigned)(L >> 3) * (LDS_ROW * 4) + (unsigned)(L & 7) * 16;
#pragma unroll
        for (int j = 0; j < 4; ++j) {
            unsigned long long ga = (unsigned long long)(uintptr_t)(g0 + (size_t)(j * 4) * KTOT);
            unsigned la = lo + (unsigned)(j * 4) * (LDS_ROW * 4);
            asm volatile("global_load_async_to_lds_b128 %0, %1, off"
                         :: "v"(la), "v"(ga) : "memory");
        }
    };

    v8f acc0 = {0.f, 0.f, 0.f, 0.f, 0.f, 0.f, 0.f, 0.f};   // rows 0..15
    v8f acc1 = {0.f, 0.f, 0.f, 0.f, 0.f, 0.f, 0.f, 0.f};   // rows 16..31

    stage_load(0, 0);
    const v2f* a2 = (const v2f*)atp;   // ATp2[kp*32 + row]

    for (int s = 0; s < NSTAGE; ++s) {
        const int p = s & 1;
        if (s + 1 < NSTAGE) {
            stage_load(s + 1, p ^ 1);                        // prefetch next stage
            asm volatile("s_wait_asynccnt 0x4" ::: "memory"); // this stage's 4 done
        } else {
            asm volatile("s_wait_asynccnt 0x0" ::: "memory");
        }
        const int kg0 = kbase + s * KSTEP;
        // B-frag (4x16 f32): lane = o, VGPR0/1 = k rows {kb,kb+1} (lanes 0-15)
        // and {kb+2,kb+3} (lanes 16-31) -> one ds_load_b64 per k-step.
        const float* bbuf = &lds[p * TILE_FLOATS + lane15 * LDS_ROW + 2 * half];
#pragma unroll
        for (int q = 0; q < KSTEP / 4; ++q) {
            v2f bf  = *(const v2f*)(bbuf + 4 * q);
            int kp0 = ((kg0 + 4 * q) >> 1) + half;
            v2f a0  = a2[kp0 * 32 + lane15];        // A-frag rows 0..15
            v2f a1  = a2[kp0 * 32 + 16 + lane15];   // A-frag rows 16..31
            acc0 = __builtin_amdgcn_wmma_f32_16x16x4_f32(
                       false, a0, false, bf, (short)0, acc0, false, false);
            acc1 = __builtin_amdgcn_wmma_f32_16x16x4_f32(
                       false, a1, false, bf, (short)0, acc1, false, false);
        }
    }

    // D layout: VGPR v -> row v (lanes 0-15) / v+8 (lanes 16-31), col = lane&15
    float* pwc = pw + (size_t)chunk * (BATCH * OUT_F) + otile * 16 + lane15;
#pragma unroll
    for (int v = 0; v < 8; ++v) {
        int rr = v + 8 * half;
        pwc[(size_t)rr * OUT_F]        = acc0[v];
        pwc[(size_t)(rr + 16) * OUT_F] = acc1[v];
    }
}

// ---------------------------------------------------------------------------
// Kernel 3: deterministic fixed-order reduction over the 32 k-chunk partials.
// ---------------------------------------------------------------------------
__global__ void reduce_pw(const float* __restrict__ pw, float* __restrict__ out) {
    int t = blockIdx.x * blockDim.x + threadIdx.x;   // < BATCH*OUT_F
    float s = 0.0f;
#pragma unroll
    for (int c = 0; c < NCHUNK; ++c) s += pw[(size_t)c * (BATCH * OUT_F) + t];
    out[t] = s;
}

extern "C" void kernel_launch(void* const* d_in, const int* in_sizes, int n_in,
                              void* d_out, int out_size, void* d_ws, size_t ws_size,
                              hipStream_t stream) {
    (void)in_sizes; (void)n_in; (void)out_size; (void)ws_size;
    const float* x = (const float*)d_in[0];   // [32, 1024]
    const float* w = (const float*)d_in[1];   // [1024, 1024, 97]
    float* out = (float*)d_out;               // [32, 1024]

    float* atp = (float*)d_ws;                         // 12.7 MB dense basis
    float* pw  = atp + (size_t)KTOT * BATCH;           // 4 MB chunk partials

    build_A<<<(KTOT * BATCH) / 256, 256, 0, stream>>>(x, atp);
    gemm_part<<<64 * NCHUNK, 32, 0, stream>>>(w, atp, pw);
    reduce_pw<<<(BATCH * OUT_F) / 256, 256, 0, stream>>>(pw, out);
}